// LinearFilterKernel_23905787969545
// MI455X (gfx1250) — compile-verified
//
#include <hip/hip_runtime.h>

// Problem constants (from reference)
#define BATCH 4
#define ST    9
#define IN_C  3
#define OUT_C 3
#define Hh    384
#define Ww    384
#define OH    128
#define OW    128
#define Kk    9
#define KK    81            // K*K taps
#define CK    243           // IN_C * KK (reduction length)
#define CKP   244           // padded to multiple of 4 (61 WMMA steps)
#define QT    4             // ow pixels per wave
#define OWG   (OW / QT)     // 32
#define ISTR  (OH * OW * KK)  // weights stride between i-planes = 1327104

typedef __attribute__((ext_vector_type(2))) float v2f;
typedef __attribute__((ext_vector_type(8))) float v8f;

static __device__ __forceinline__ v8f wmma_step(v2f a, v2f b, v8f c) {
    return __builtin_amdgcn_wmma_f32_16x16x4_f32(
        /*neg_a=*/false, a, /*neg_b=*/false, b,
        /*c_mod=*/(short)0, c, /*reuse_a=*/false, /*reuse_b=*/false);
}

// One wave per (st, oh, ow-group-of-4).
// A (LDS, 16x244): rows m = 4*q + b -> patch[b, pixel q, k]   (k = i*81 + ky*9 + kx)
// B (global, streamed): rows n = 4*o + q -> weight[o, pixel q, k]; lane L reads its
//   own row n = L%16 at k = 4s + {0,1 | 2,3}, i.e. a contiguous per-lane stream.
// D = A x B via V_WMMA_F32_16X16X4_F32; entries with q_row == q_col are the outputs.
__global__ void lf_filter_wmma_kernel(const float* __restrict__ lf,
                                      const float* __restrict__ wts,
                                      const float* __restrict__ biases,
                                      float* __restrict__ out) {
    __shared__ __align__(16) float Apack[16 * CKP];

    const int lane = threadIdx.x;          // 0..31
    const int lin  = blockIdx.x;           // st*4096 + oh*32 + og
    const int og   = lin & (OWG - 1);
    const int oh   = (lin >> 5) & (OH - 1);
    const int st   = lin >> 12;            // 0..8
    const int ow0  = og * QT;

    // ---------------- stage A: one (m,i,ky) 9-tap row per lane ----------------
    // 16 m-rows * 3 i * 9 ky = 432 rows total
    for (int j = 0; j < 14; ++j) {
        const int r = lane + 32 * j;
        if (r < 432) {
            const int m   = r / 27;
            const int rem = r - m * 27;
            const int i   = rem / 9;
            const int ky  = rem - i * 9;
            const int b   = m & 3;
            const int q   = m >> 2;
            const int y   = oh * 3 + ky - 3;
            const int x0  = (ow0 + q) * 3 - 3;
            const bool yok = (unsigned)y < (unsigned)Hh;
            float* dst = &Apack[m * CKP + i * KK + ky * Kk];
            const float* src = lf + ((size_t)((b * ST + st) * IN_C + i) * Hh + (yok ? y : 0)) * Ww;
            if (yok && x0 >= 0 && x0 <= Ww - Kk) {         // interior fast path
                #pragma unroll
                for (int kx = 0; kx < Kk; ++kx) dst[kx] = src[x0 + kx];
            } else {                                        // image border: per-tap guard
                #pragma unroll
                for (int kx = 0; kx < Kk; ++kx) {
                    const int x = x0 + kx;
                    float v = 0.0f;
                    if (yok && (unsigned)x < (unsigned)Ww) v = src[x];
                    dst[kx] = v;
                }
            }
        }
    }
    if (lane < 16) Apack[lane * CKP + CK] = 0.0f;   // K tail column (k = 243)

    // ---------------- B: per-lane weight row stream (no LDS) ----------------
    const int  n    = lane & 15;             // D column this lane owns
    const int  nc   = (n < 12) ? n : 11;     // clamp unused cols to a valid row
    const int  o    = nc >> 2;
    const int  q    = nc & 3;
    const bool hi   = lane >= 16;
    const int  koff = hi ? 2 : 0;
    const float* rb0 = wts + (size_t)((((st * OUT_C + o) * IN_C) * OH + oh) * OW + (ow0 + q)) * KK;
    const float* rb1 = rb0 + (ISTR - KK);        // rb1[k] valid for k in [81,162)
    const float* rb2 = rb0 + 2 * (ISTR - KK);    // rb2[k] valid for k in [162,243)

    __builtin_prefetch(rb0 + koff,       0, 0);  // global_prefetch_b8 on segment heads
    __builtin_prefetch(rb1 + 81 + koff,  0, 0);
    __builtin_prefetch(rb2 + 162 + koff, 0, 0);

    __syncthreads();

    // ---------------- WMMA reduction, K = 244 in 61 steps of 4 ----------------
    const float* ap = &Apack[n * CKP + koff];    // A row = lane%16 (all 16 valid)
    v8f acc = {};

    #pragma unroll
    for (int s = 0; s < 20; ++s) {               // k in i=0 plane for both halves
        v2f a = *(const v2f*)(ap + 4 * s);
        v2f b; b.x = rb0[4 * s + koff]; b.y = rb0[4 * s + koff + 1];
        acc = wmma_step(a, b, acc);
    }
    {   // s = 20: low half pair (80,81) straddles i=0 -> i=1
        v2f a = *(const v2f*)(ap + 80);
        const float* px = hi ? (rb1 + 82) : (rb0 + 80);
        const float* py = hi ? (rb1 + 83) : (rb1 + 81);
        v2f b; b.x = *px; b.y = *py;
        acc = wmma_step(a, b, acc);
    }
    #pragma unroll
    for (int s = 21; s < 40; ++s) {              // k in i=1 plane for both halves
        v2f a = *(const v2f*)(ap + 4 * s);
        v2f b; b.x = rb1[4 * s + koff]; b.y = rb1[4 * s + koff + 1];
        acc = wmma_step(a, b, acc);
    }
    {   // s = 40: high half pair (162,163) is in i=2, low pair (160,161) still i=1
        v2f a = *(const v2f*)(ap + 160);
        const float* px = hi ? (rb2 + 162) : (rb1 + 160);
        const float* py = hi ? (rb2 + 163) : (rb1 + 161);
        v2f b; b.x = *px; b.y = *py;
        acc = wmma_step(a, b, acc);
    }
    #pragma unroll
    for (int s = 41; s < 60; ++s) {              // k in i=2 plane for both halves
        v2f a = *(const v2f*)(ap + 4 * s);
        v2f b; b.x = rb2[4 * s + koff]; b.y = rb2[4 * s + koff + 1];
        acc = wmma_step(a, b, acc);
    }
    {   // s = 60: K tail; k = 243 does not exist -> multiply against literal 0
        v2f a = *(const v2f*)(ap + 240);
        v2f b;
        b.x = hi ? rb2[242] : rb2[240];
        b.y = hi ? 0.0f     : rb2[241];
        acc = wmma_step(a, b, acc);
    }

    // ---------------- extract diagonal (q_row == q_col), bias, clamp ----------------
    // D layout: VGPR r, lanes 0-15 -> M=r, lanes 16-31 -> M=r+8, N = lane%16.
    // Useful entries: M = 4*qn + b for this lane's column n = 4*oo + qn.
    const int qn = n & 3;
    const int oo = n >> 2;
    if (oo < OUT_C && ((qn >= 2) == hi)) {
        const float bv = biases[((st * OUT_C + oo) * OH + oh) * OW + (ow0 + qn)];
        float vals[4];
        if (qn & 1) { vals[0] = acc[4]; vals[1] = acc[5]; vals[2] = acc[6]; vals[3] = acc[7]; }
        else        { vals[0] = acc[0]; vals[1] = acc[1]; vals[2] = acc[2]; vals[3] = acc[3]; }
        #pragma unroll
        for (int bb = 0; bb < BATCH; ++bb) {
            float v = vals[bb] + bv;
            v = v < 0.0f ? 0.0f : (v > 1.0f ? 1.0f : v);
            out[(((bb * ST + st) * OUT_C + oo) * OH + oh) * OW + (ow0 + qn)] = v;
        }
    }
}

extern "C" void kernel_launch(void* const* d_in, const int* in_sizes, int n_in,
                              void* d_out, int out_size, void* d_ws, size_t ws_size,
                              hipStream_t stream) {
    (void)in_sizes; (void)n_in; (void)out_size; (void)d_ws; (void)ws_size;
    const float* lf     = (const float*)d_in[0];
    const float* wts    = (const float*)d_in[1];
    const float* biases = (const float*)d_in[2];
    float* out          = (float*)d_out;

    const int nblocks = ST * OH * OWG;   // 9 * 128 * 32 = 36864 waves
    lf_filter_wmma_kernel<<<nblocks, 32, 0, stream>>>(lf, wts, biases, out);
}